// NNP_27779848470903
// MI455X (gfx1250) — compile-verified
//
#include <hip/hip_runtime.h>

// Problem sizes (fixed by the reference)
#define NA 512      // atoms
#define DD 128      // descriptor dim
#define HH 128      // hidden dim
#define NS 4        // species
#define K3 (3*NA)   // 1536 force components

typedef __attribute__((ext_vector_type(2))) float v2f;
typedef __attribute__((ext_vector_type(4))) float v4f;
typedef __attribute__((ext_vector_type(8))) float v8f;

static __device__ __forceinline__ v8f wmma_f32(v2f a, v2f b, v8f c) {
  // V_WMMA_F32_16X16X4_F32 : D(16x16) = A(16x4) * B(4x16) + C
  return __builtin_amdgcn_wmma_f32_16x16x4_f32(false, a, false, b, (short)0, c,
                                               false, false);
}

// ---------------------------------------------------------------------------
// Kernel 1: per-(16-atom tile, species) MLP forward + backward to produce
//   E[n]  (selected by species) and G[n][d] = dE/d desc[n][d].
// Block = 256 threads = 8 waves; wave w owns the 16-row output tile [16w,16w+16).
// WMMA operand mapping (wave32, 32-bit layouts from CDNA5 ISA 7.12.2):
//   lane = 16*half + l ;  A[l + m0][kk + 2*half + j] in a.{x,y}
//   B[kk + 2*half + j][l] in b.{x,y} ;  D VGPR g = row (m0 + g + 8*half), col l
// ---------------------------------------------------------------------------
__global__ __launch_bounds__(256) void mlp_grad_kernel(
    const float* __restrict__ desc, const float* __restrict__ W0,
    const float* __restrict__ b0, const float* __restrict__ W1,
    const float* __restrict__ b1, const float* __restrict__ W2,
    const float* __restrict__ b2, const int* __restrict__ species,
    float* __restrict__ Gout, float* __restrict__ Eout) {
  __shared__ float bufA[HH * 16];  // h0, later overwritten (in-place) with t0
  __shared__ float bufB[HH * 16];  // t1 = (1-h1^2) * w2
  __shared__ float eacc[16];       // per-atom energy accumulator

  const int n0 = blockIdx.x * 16;   // atom tile base
  const int s  = blockIdx.y;        // species handled by this block
  const int wave = threadIdx.x >> 5;
  const int lane = threadIdx.x & 31;
  const int half = lane >> 4;
  const int l    = lane & 15;
  const int m0   = wave * 16;       // output-row tile of this wave

  const int myspec = species[n0 + l];  // species of the atom in this lane's column

  const float* w0 = W0 + (size_t)s * HH * DD;
  const float* w1 = W1 + (size_t)s * HH * HH;
  const float* w2 = W2 + (size_t)s * HH;

  if (threadIdx.x < 16) eacc[threadIdx.x] = 0.f;

  // ---- layer 0: h0 = tanh(W0 @ desc^T + b0)  (128 x 16) -------------------
  v8f c0 = {};
  {
    const float* arow = w0 + (size_t)(m0 + l) * DD + 2 * half;
    const float* brow = desc + (size_t)(n0 + l) * DD + 2 * half;
#pragma unroll 4
    for (int kk = 0; kk < DD; kk += 4) {
      v2f a = *(const v2f*)(arow + kk);
      v2f b = *(const v2f*)(brow + kk);
      c0 = wmma_f32(a, b, c0);
    }
  }
#pragma unroll
  for (int g = 0; g < 8; ++g) {
    const int m = m0 + g + 8 * half;
    const float h = tanhf(c0[g] + b0[s * HH + m]);
    bufA[m * 16 + l] = h;
  }
  __syncthreads();  // h0 visible to all waves; eacc zeroed

  // ---- layer 1: h1 = tanh(W1 @ h0 + b1); e += w2.h1 ; t1 = (1-h1^2)*w2 ----
  v8f c1 = {};
  {
    const float* arow = w1 + (size_t)(m0 + l) * HH + 2 * half;
#pragma unroll 4
    for (int kk = 0; kk < HH; kk += 4) {
      v2f a = *(const v2f*)(arow + kk);
      v2f b;
      b.x = bufA[(kk + 2 * half) * 16 + l];
      b.y = bufA[(kk + 2 * half + 1) * 16 + l];
      c1 = wmma_f32(a, b, c1);
    }
  }
  float epart = 0.f;
#pragma unroll
  for (int g = 0; g < 8; ++g) {
    const int m = m0 + g + 8 * half;
    const float h = tanhf(c1[g] + b1[s * HH + m]);
    const float w2m = w2[m];
    epart += w2m * h;
    bufB[m * 16 + l] = (1.f - h * h) * w2m;  // t1
  }
  atomicAdd(&eacc[l], epart);  // ds_add_f32; 16 contributors per atom column
  __syncthreads();             // t1 + eacc complete

  if (wave == 0 && half == 0 && myspec == s) Eout[n0 + l] = eacc[l] + b2[s];

  // ---- backward through layer 1: d0 = W1^T @ t1 ; t0 = (1-h0^2)*d0 --------
  v8f cd = {};
#pragma unroll 4
  for (int kk = 0; kk < HH; kk += 4) {
    const int r0 = kk + 2 * half;
    v2f a;  // A = W1^T : A[m0+l][r] = W1[r][m0+l]
    a.x = w1[(size_t)r0 * HH + m0 + l];
    a.y = w1[(size_t)(r0 + 1) * HH + m0 + l];
    v2f b;
    b.x = bufB[r0 * 16 + l];
    b.y = bufB[(r0 + 1) * 16 + l];
    cd = wmma_f32(a, b, cd);
  }
#pragma unroll
  for (int g = 0; g < 8; ++g) {  // in-place: this wave only touches its own rows
    const int m = m0 + g + 8 * half;
    const float h0v = bufA[m * 16 + l];
    bufA[m * 16 + l] = (1.f - h0v * h0v) * cd[g];  // t0
  }
  __syncthreads();  // t0 visible to all waves

  // ---- G = W0^T @ t0  (128 x 16) ------------------------------------------
  v8f cgrad = {};
#pragma unroll 4
  for (int kk = 0; kk < HH; kk += 4) {
    const int r0 = kk + 2 * half;
    v2f a;  // A = W0^T : A[m0+l][r] = W0[r][m0+l]
    a.x = w0[(size_t)r0 * DD + m0 + l];
    a.y = w0[(size_t)(r0 + 1) * DD + m0 + l];
    v2f b;
    b.x = bufA[r0 * 16 + l];
    b.y = bufA[(r0 + 1) * 16 + l];
    cgrad = wmma_f32(a, b, cgrad);
  }
  // species routing: only the block whose s matches this atom writes its column
  if (myspec == s) {
#pragma unroll
    for (int g = 0; g < 8; ++g)
      Gout[(size_t)(n0 + l) * DD + m0 + g + 8 * half] = cgrad[g];
  }
}

// ---------------------------------------------------------------------------
// Kernel 2: forces[k] = sum_n dot(coord_grads[n][k][:], G[n][:])
// 403 MB streamed once -> pure HBM roofline. Non-temporal loads on the
// streamed tensor keep G (256 KB) resident in L2 across all 1536 blocks.
// ---------------------------------------------------------------------------
__global__ __launch_bounds__(256) void forces_kernel(
    const float* __restrict__ cg, const float* __restrict__ G,
    float* __restrict__ out) {
  __shared__ float red[256];
  const int k = blockIdx.x;  // 0..1535
  const int t = threadIdx.x;
  const int dq = (t & 31) << 2;  // float4 slot within the 128-dim dot
  const int ng = t >> 5;         // 8 n-phases
  float acc = 0.f;
  for (int n = ng; n < NA; n += 8) {
    const v4f a =
        __builtin_nontemporal_load((const v4f*)(cg + ((size_t)n * K3 + k) * DD + dq));
    const v4f g4 = *(const v4f*)(G + (size_t)n * DD + dq);
    acc += a.x * g4.x + a.y * g4.y + a.z * g4.z + a.w * g4.w;
  }
  red[t] = acc;
  __syncthreads();
#pragma unroll
  for (int off = 128; off > 0; off >>= 1) {
    if (t < off) red[t] += red[t + off];
    __syncthreads();
  }
  if (t == 0) out[1 + k] = red[0];
}

// ---------------------------------------------------------------------------
// Kernel 3: blocks 0..5 -> stress rows; block 6 -> deterministic E reduction
// ---------------------------------------------------------------------------
__global__ __launch_bounds__(256) void stress_energy_kernel(
    const float* __restrict__ sg, const float* __restrict__ G,
    const float* __restrict__ E, const float* __restrict__ volume,
    float* __restrict__ out) {
  __shared__ float red[256];
  const int t = threadIdx.x;
  float acc = 0.f;
  if (blockIdx.x < 6) {
    const int j = blockIdx.x;
    const int dq = (t & 31) << 2;
    const int ng = t >> 5;
    for (int n = ng; n < NA; n += 8) {
      const v4f a = *(const v4f*)(sg + ((size_t)n * 6 + j) * DD + dq);
      const v4f g4 = *(const v4f*)(G + (size_t)n * DD + dq);
      acc += a.x * g4.x + a.y * g4.y + a.z * g4.z + a.w * g4.w;
    }
  } else {
    acc = E[t] + E[t + 256];
  }
  red[t] = acc;
  __syncthreads();
#pragma unroll
  for (int off = 128; off > 0; off >>= 1) {
    if (t < off) red[t] += red[t + off];
    __syncthreads();
  }
  if (t == 0) {
    if (blockIdx.x < 6)
      out[1 + K3 + blockIdx.x] = -red[0] / volume[0];
    else
      out[0] = red[0];
  }
}

extern "C" void kernel_launch(void* const* d_in, const int* in_sizes, int n_in,
                              void* d_out, int out_size, void* d_ws,
                              size_t ws_size, hipStream_t stream) {
  (void)in_sizes; (void)n_in; (void)out_size; (void)ws_size;
  const float* desc = (const float*)d_in[0];   // (512,128)
  const float* cg   = (const float*)d_in[1];   // (512,1536,128)
  const float* sg   = (const float*)d_in[2];   // (512,6,128)
  const float* W0   = (const float*)d_in[3];   // (4,128,128)
  const float* b0   = (const float*)d_in[4];   // (4,128)
  const float* W1   = (const float*)d_in[5];   // (4,128,128)
  const float* b1   = (const float*)d_in[6];   // (4,128)
  const float* W2   = (const float*)d_in[7];   // (4,1,128)
  const float* b2   = (const float*)d_in[8];   // (4,1)
  const float* vol  = (const float*)d_in[9];   // (1,)
  const int* species = (const int*)d_in[10];   // (512,)
  float* out = (float*)d_out;                  // 1 + 1536 + 6

  float* Gws = (float*)d_ws;                   // 512*128 floats
  float* Ews = Gws + (size_t)NA * DD;          // 512 floats

  dim3 grid1(NA / 16, NS);
  mlp_grad_kernel<<<grid1, 256, 0, stream>>>(desc, W0, b0, W1, b1, W2, b2,
                                             species, Gws, Ews);
  forces_kernel<<<K3, 256, 0, stream>>>(cg, Gws, out);
  stress_energy_kernel<<<7, 256, 0, stream>>>(sg, Gws, Ews, vol, out);
}